// NCF_AKBIPS_Exp_2911987826854
// MI455X (gfx1250) — compile-verified
//
#include <hip/hip_runtime.h>
#include <hip/hip_bf16.h>
#include <math.h>

#define NV 8192
#define FD 16
#define GAMMA 0.1f

typedef __attribute__((ext_vector_type(2))) float v2f;
typedef __attribute__((ext_vector_type(8))) float v8f;

// feat[row][col] = col<8 ? W[idx[row,0]][col] : H[idx[row,1]][col-8]
__device__ __forceinline__ float feat_val(const int* __restrict__ idx,
                                          const float* __restrict__ W,
                                          const float* __restrict__ H,
                                          int row, int col) {
  if (col < 8) return W[(size_t)idx[2 * row] * 8 + col];
  return H[(size_t)idx[2 * row + 1] * 8 + (col - 8)];
}

// ---------------- Stage 1: per-block min/max of featx and featy ----------------
__global__ void __launch_bounds__(256) minmax_stage1(const int* __restrict__ x,
                                                     const int* __restrict__ y,
                                                     const float* __restrict__ W,
                                                     const float* __restrict__ H,
                                                     float* __restrict__ partials) {
  const int tid = threadIdx.x;
  const int gid = blockIdx.x * blockDim.x + tid;
  const int stride = gridDim.x * blockDim.x;
  float mnx = INFINITY, mxx = -INFINITY, mny = INFINITY, mxy = -INFINITY;
  for (int e = gid; e < NV * FD; e += stride) {
    int row = e >> 4, col = e & 15;
    float vx = feat_val(x, W, H, row, col);
    float vy = feat_val(y, W, H, row, col);
    mnx = fminf(mnx, vx); mxx = fmaxf(mxx, vx);
    mny = fminf(mny, vy); mxy = fmaxf(mxy, vy);
  }
  __shared__ float s[256];
  // minX -> partials[0*128+b], maxX -> [1*128+b], minY -> [2*128+b], maxY -> [3*128+b]
  float vals[4] = {mnx, mxx, mny, mxy};
#pragma unroll
  for (int q = 0; q < 4; ++q) {
    s[tid] = vals[q];
    __syncthreads();
    for (int o = 128; o > 0; o >>= 1) {
      if (tid < o) s[tid] = (q & 1) ? fmaxf(s[tid], s[tid + o]) : fminf(s[tid], s[tid + o]);
      __syncthreads();
    }
    if (tid == 0) partials[q * 128 + blockIdx.x] = s[0];
    __syncthreads();
  }
}

// ---------------- Stage 2: final reduce + scalar shifts ----------------
__global__ void __launch_bounds__(128) minmax_stage2(const float* __restrict__ partials,
                                                     float* __restrict__ scalars) {
  __shared__ float s[128];
  const int t = threadIdx.x;
  __shared__ float r[4];
#pragma unroll
  for (int q = 0; q < 4; ++q) {
    s[t] = partials[q * 128 + t];
    __syncthreads();
    for (int o = 64; o > 0; o >>= 1) {
      if (t < o) s[t] = (q & 1) ? fmaxf(s[t], s[t + o]) : fminf(s[t], s[t + o]);
      __syncthreads();
    }
    if (t == 0) r[q] = s[0];
    __syncthreads();
  }
  if (t == 0) {
    // reference: feat = feat - f_min/(f_max - f_min)   (scalar shift)
    scalars[0] = r[0] / (r[1] - r[0]);  // cx
    scalars[1] = r[2] / (r[3] - r[2]);  // cy
  }
}

// ---------------- Build fx, fy (8192x16) and row norms ----------------
__global__ void __launch_bounds__(256) build_feats(const int* __restrict__ x,
                                                   const int* __restrict__ y,
                                                   const float* __restrict__ W,
                                                   const float* __restrict__ H,
                                                   const float* __restrict__ scalars,
                                                   float* __restrict__ fx, float* __restrict__ fy,
                                                   float* __restrict__ nx, float* __restrict__ ny) {
  const int i = blockIdx.x * blockDim.x + threadIdx.x;
  if (i >= NV) return;
  const float cx = scalars[0], cy = scalars[1];
  float sx = 0.f, sy = 0.f;
#pragma unroll
  for (int c = 0; c < FD; ++c) {
    float vx = (feat_val(x, W, H, i, c) - cx) * (1.0f / 16.0f);
    float vy = (feat_val(y, W, H, i, c) - cy) * (1.0f / 16.0f);
    fx[i * FD + c] = vx;
    fy[i * FD + c] = vy;
    sx += vx * vx;
    sy += vy * vy;
  }
  nx[i] = sx;
  ny[i] = sy;
}

// ---------------- Main kernel: K = exp(-gamma * dist) via V_WMMA_F32_16X16X4_F32 ----------------
// One 16x16 output tile per wave32; 8 waves per 256-thread block.
// A = fy tile (16x4 per step), B = fx tile transposed (4x16 per step); both load as
// two consecutive fp32 per lane from row-major fx/fy (ISA 7.12.2 32-bit layouts).
// Epilogue uses raw v_sqrt_f32 / v_exp_f32 so the kernel stays under the
// 256MB-store bandwidth roofline instead of going VALU-bound on IEEE sqrt refinement.
__global__ void __launch_bounds__(256) gemm_exp_kernel(const float* __restrict__ fx,
                                                       const float* __restrict__ fy,
                                                       const float* __restrict__ nx,
                                                       const float* __restrict__ ny,
                                                       float* __restrict__ out) {
  const int wave = blockIdx.x * (blockDim.x >> 5) + (threadIdx.x >> 5);
  const int lane = threadIdx.x & 31;
  const int hl = lane >> 4;   // lane half selects K-pair (operands) / M-half (result)
  const int mn = lane & 15;   // A: row M ; B: column N
  const int tileI = wave >> 9;    // / (8192/16)
  const int tileJ = wave & 511;
  const int i0 = tileI * 16;
  const int j0 = tileJ * 16;

  const float* __restrict__ Arow = fy + (size_t)(i0 + mn) * FD;  // A[M][k]
  const float* __restrict__ Brow = fx + (size_t)(j0 + mn) * FD;  // B[k][N] = fx[j0+N][k]

  v8f acc = {0.f, 0.f, 0.f, 0.f, 0.f, 0.f, 0.f, 0.f};
#pragma unroll
  for (int kk = 0; kk < 4; ++kk) {
    const int k0 = kk * 4 + hl * 2;
    v2f a = *(const v2f*)(Arow + k0);
    v2f b = *(const v2f*)(Brow + k0);
    // D = A(16x4,f32) x B(4x16,f32) + C(16x16,f32)
    acc = __builtin_amdgcn_wmma_f32_16x16x4_f32(false, a, false, b, (short)0, acc,
                                                false, false);
  }

  const float nxj = nx[j0 + mn];
  const float* __restrict__ nyp = ny + i0 + hl * 8;  // 8 consecutive rows for this half
#pragma unroll
  for (int r = 0; r < 8; ++r) {
    const int i = i0 + r + hl * 8;           // C/D layout: VGPR r, lane half -> row
    const float d2 = nyp[r] + nxj - 2.0f * acc[r];
    const float d = __builtin_amdgcn_sqrtf(fmaxf(d2, 0.0f));  // raw v_sqrt_f32
    out[(size_t)i * NV + (j0 + mn)] = __expf(-GAMMA * d);     // v_exp_f32
  }
}

// ---------------- pred (tiny MLP) and w (MF dot) ----------------
__global__ void __launch_bounds__(256) pred_w_kernel(const int* __restrict__ x,
                                                     const float* __restrict__ ncf_u,
                                                     const float* __restrict__ ncf_i,
                                                     const float* __restrict__ l1w,   // (8,16)
                                                     const float* __restrict__ l1b,   // (8)
                                                     const float* __restrict__ l2w,   // (1,8)
                                                     const float* __restrict__ mf_u,
                                                     const float* __restrict__ mf_i,
                                                     float* __restrict__ out) {
  const int i = blockIdx.x * blockDim.x + threadIdx.x;
  if (i >= NV) return;
  const int u = x[2 * i];
  const int it = x[2 * i + 1];
  float z[16];
#pragma unroll
  for (int c = 0; c < 8; ++c) z[c] = ncf_u[(size_t)u * 8 + c];
#pragma unroll
  for (int c = 0; c < 8; ++c) z[8 + c] = ncf_i[(size_t)it * 8 + c];

  float pred = 0.f;
#pragma unroll
  for (int k = 0; k < 8; ++k) {
    float h = l1b[k];
#pragma unroll
    for (int c = 0; c < 16; ++c) h += z[c] * l1w[k * 16 + c];
    pred += fmaxf(h, 0.f) * l2w[k];
  }
  out[(size_t)NV * NV + i] = pred;

  float wv = 0.f;
#pragma unroll
  for (int c = 0; c < 8; ++c) wv += mf_u[(size_t)u * 8 + c] * mf_i[(size_t)it * 8 + c];
  out[(size_t)NV * NV + NV + i] = wv;
}

extern "C" void kernel_launch(void* const* d_in, const int* in_sizes, int n_in,
                              void* d_out, int out_size, void* d_ws, size_t ws_size,
                              hipStream_t stream) {
  const int*   x     = (const int*)d_in[0];
  const int*   y     = (const int*)d_in[1];
  const float* W     = (const float*)d_in[2];
  const float* H     = (const float*)d_in[3];
  const float* ncf_u = (const float*)d_in[4];
  const float* ncf_i = (const float*)d_in[5];
  const float* l1w   = (const float*)d_in[6];
  const float* l1b   = (const float*)d_in[7];
  const float* l2w   = (const float*)d_in[8];
  const float* mf_u  = (const float*)d_in[9];
  const float* mf_i  = (const float*)d_in[10];
  float* out = (float*)d_out;

  // workspace layout (floats): fx | fy | nx | ny | partials(512) | scalars(2)
  float* ws       = (float*)d_ws;
  float* fx       = ws;                    // 131072
  float* fy       = fx + NV * FD;          // 131072
  float* nx       = fy + NV * FD;          // 8192
  float* ny       = nx + NV;               // 8192
  float* partials = ny + NV;               // 512
  float* scalars  = partials + 512;        // 2

  minmax_stage1<<<128, 256, 0, stream>>>(x, y, W, H, partials);
  minmax_stage2<<<1, 128, 0, stream>>>(partials, scalars);
  build_feats<<<(NV + 255) / 256, 256, 0, stream>>>(x, y, W, H, scalars, fx, fy, nx, ny);

  // 512x512 tiles of 16x16, one tile per wave32, 8 waves per block
  const int totalWaves = (NV / 16) * (NV / 16);   // 262144
  gemm_exp_kernel<<<totalWaves / 8, 256, 0, stream>>>(fx, fy, nx, ny, out);

  pred_w_kernel<<<(NV + 255) / 256, 256, 0, stream>>>(x, ncf_u, ncf_i, l1w, l1b, l2w,
                                                      mf_u, mf_i, out);
}